// AttentionBlock_48996986912955
// MI455X (gfx1250) — compile-verified
//
#include <hip/hip_runtime.h>
#include <math.h>

#define IN_CH   768
#define OUT_CH  48
#define BATCH   32
#define SPATIAL 3136   // 56*56

typedef __attribute__((ext_vector_type(2))) float v2f;
typedef __attribute__((ext_vector_type(8))) float v8f;

// ---------------------------------------------------------------------------
// Kernel 1: fused global avg + max pool. One wave32 per (b,c) row of 3136
// contiguous floats. float4 loads -> global_load_b128; wave32 butterfly
// reduction via __shfl_xor.
// ---------------------------------------------------------------------------
__global__ void __launch_bounds__(256)
pool_kernel(const float* __restrict__ x,
            float* __restrict__ avg_out,
            float* __restrict__ max_out) {
  const int lane = threadIdx.x & 31;
  const int wave = threadIdx.x >> 5;          // 8 waves per block
  const int row  = blockIdx.x * 8 + wave;     // row = b*IN_CH + c
  const float4* xr = (const float4*)(x + (size_t)row * SPATIAL);

  float s = 0.0f;
  float m = -3.402823466e38f;
  for (int i = lane; i < SPATIAL / 4; i += 32) {
    float4 v = xr[i];
    s += v.x + v.y + v.z + v.w;
    m = fmaxf(m, fmaxf(fmaxf(v.x, v.y), fmaxf(v.z, v.w)));
  }
  for (int off = 16; off > 0; off >>= 1) {
    s += __shfl_xor(s, off, 32);
    m = fmaxf(m, __shfl_xor(m, off, 32));
  }
  if (lane == 0) {
    avg_out[row] = s * (1.0f / (float)SPATIAL);
    max_out[row] = m;
  }
}

// ---------------------------------------------------------------------------
// Kernel 2: gate = sigmoid(mlp(avg) + mlp(max)) using V_WMMA_F32_16X16X4_F32.
//
// Fragment layouts per CDNA5 ISA 7.12.2 (wave32):
//   A (16x4, MxK, 2 VGPRs/lane): lane L holds M = L&15,
//       v0 = A[M][k0 + 2*(L>>4)], v1 = A[M][k0 + 2*(L>>4) + 1]
//   B (4x16, KxN, 2 VGPRs/lane): lane L holds N = L&15, same K striping
//   C/D (16x16, 8 VGPRs): vgpr i, lanes 0-15 -> M=i, lanes 16-31 -> M=i+8,
//       N = lane & 15
// One 384-thread block (12 waves); no divergence (EXEC all-1s for WMMA).
// ---------------------------------------------------------------------------
__global__ void __launch_bounds__(384)
gate_kernel(const float* __restrict__ pool_avg,   // [32][768]
            const float* __restrict__ pool_max,   // [32][768]
            const float* __restrict__ w1,         // [48][768]
            const float* __restrict__ w2,         // [768][48]
            float* __restrict__ gate) {           // [32][768]
  __shared__ float h_sh[2][BATCH][OUT_CH];        // relu(v @ w1^T), 12 KB

  const int lane = threadIdx.x & 31;
  const int wave = threadIdx.x >> 5;              // 0..11
  const int half = lane >> 4;                     // 0 or 1 (K striping)
  const int l16  = lane & 15;

  // ---- Phase A: h[which] = relu(v @ w1^T), [32]x[768] * [768]x[48] ----
  {
    const int which = wave / 6;                   // 0=avg, 1=max
    const int rem   = wave % 6;
    const int mt    = rem / 3;                    // M tile: 0..1
    const int nt    = rem % 3;                    // N tile: 0..2
    const float* v  = which ? pool_max : pool_avg;
    const int m = mt * 16 + l16;
    const int n = nt * 16 + l16;

    v8f acc = {};
    for (int k0 = 0; k0 < IN_CH; k0 += 4) {
      const int k = k0 + 2 * half;
      v2f a, b;
      a.x = v[m * IN_CH + k];                     // A = v  (B x C)
      a.y = v[m * IN_CH + k + 1];
      b.x = w1[n * IN_CH + k];                    // B[k][n] = w1[n][k]
      b.y = w1[n * IN_CH + k + 1];
      acc = __builtin_amdgcn_wmma_f32_16x16x4_f32(
          /*neg_a=*/false, a, /*neg_b=*/false, b,
          /*c_mod=*/(short)0, acc, /*reuse_a=*/false, /*reuse_b=*/false);
    }
    for (int i = 0; i < 8; ++i) {
      float hv = acc[i];
      h_sh[which][mt * 16 + i + half * 8][nt * 16 + l16] = hv > 0.0f ? hv : 0.0f;
    }
  }
  __syncthreads();

  // ---- Phase B: gate = sigmoid(h_avg @ w2^T + h_max @ w2^T) ----
  // 2 M-tiles x 48 N-tiles = 96 tiles; 8 tiles per wave.
  for (int t = wave * 8; t < wave * 8 + 8; ++t) {
    const int mt = t / 48;
    const int nt = t % 48;
    const int n  = nt * 16 + l16;

    v8f acc = {};
    for (int which = 0; which < 2; ++which) {
      for (int k0 = 0; k0 < OUT_CH; k0 += 4) {
        const int k = k0 + 2 * half;
        v2f a, b;
        a.x = h_sh[which][mt * 16 + l16][k];      // A = h (B x O)
        a.y = h_sh[which][mt * 16 + l16][k + 1];
        b.x = w2[n * OUT_CH + k];                 // B[k=o][n=c] = w2[c][o]
        b.y = w2[n * OUT_CH + k + 1];
        acc = __builtin_amdgcn_wmma_f32_16x16x4_f32(
            false, a, false, b, (short)0, acc, false, false);
      }
    }
    for (int i = 0; i < 8; ++i) {
      const int m = mt * 16 + i + half * 8;
      gate[m * IN_CH + n] = 1.0f / (1.0f + __expf(-acc[i]));
    }
  }
}

// ---------------------------------------------------------------------------
// Kernel 3: out[b,c,h,w] = gate[b,c] * in[b,c,h,w], float4 vectorized.
// SPATIAL % 4 == 0 so each float4 stays within one (b,c) row.
// ---------------------------------------------------------------------------
__global__ void __launch_bounds__(256)
scale_kernel(const float* __restrict__ x,
             const float* __restrict__ gate,
             float* __restrict__ out,
             int n4) {
  const int idx = blockIdx.x * blockDim.x + threadIdx.x;
  if (idx >= n4) return;
  const int bc = (idx << 2) / SPATIAL;
  const float g = gate[bc];
  float4 v = ((const float4*)x)[idx];
  v.x *= g; v.y *= g; v.z *= g; v.w *= g;
  ((float4*)out)[idx] = v;
}

// ---------------------------------------------------------------------------
extern "C" void kernel_launch(void* const* d_in, const int* in_sizes, int n_in,
                              void* d_out, int out_size, void* d_ws, size_t ws_size,
                              hipStream_t stream) {
  const float* x  = (const float*)d_in[0];   // [32,768,56,56]
  const float* w1 = (const float*)d_in[1];   // [48,768]
  const float* w2 = (const float*)d_in[2];   // [768,48]
  float* out = (float*)d_out;

  float* ws       = (float*)d_ws;            // 3 * 24576 floats = 295 KB
  float* pool_avg = ws;
  float* pool_max = ws + BATCH * IN_CH;
  float* gate     = ws + 2 * BATCH * IN_CH;

  const int rows = BATCH * IN_CH;            // 24576 (b,c) rows
  pool_kernel<<<rows / 8, 256, 0, stream>>>(x, pool_avg, pool_max);

  gate_kernel<<<1, 384, 0, stream>>>(pool_avg, pool_max, w1, w2, gate);

  const int n4 = (BATCH * IN_CH * SPATIAL) / 4;   // 19,267,584 float4s
  scale_kernel<<<(n4 + 255) / 256, 256, 0, stream>>>(x, gate, out, n4);
}